// MambaEncoderBlock_82265803587910
// MI455X (gfx1250) — compile-verified
//
#include <hip/hip_runtime.h>
#include <hip/hip_bf16.h>

typedef float v2f_t __attribute__((ext_vector_type(2)));
typedef float v8f_t __attribute__((ext_vector_type(8)));
typedef unsigned int u32x4_t __attribute__((ext_vector_type(4)));
typedef int i32x8_t __attribute__((ext_vector_type(8)));
typedef int i32x4_t __attribute__((ext_vector_type(4)));

#define BM 128
#define BN 64
#define KC 32

#ifndef __has_builtin
#define __has_builtin(x) 0
#endif
#if __has_builtin(__builtin_amdgcn_tensor_load_to_lds) && \
    __has_builtin(__builtin_amdgcn_s_wait_tensorcnt)
#define USE_TDM 1
#else
#define USE_TDM 0
#endif

__device__ __forceinline__ float softplusf(float x) {
    return (x > 20.f) ? x : log1pf(__expf(x));
}

#if USE_TDM
// ---------------------------------------------------------------------------
// TDM 2-D tile load: global (row-major, stride0 elements of 4B) -> LDS.
// Descriptor per CDNA5 ISA ch.8: group0 = count/lds/global/type,
// group1 = data_size + pad + dims/strides, later groups zero (<=2D tile).
// tensor_dim0/1 give the in-bounds extent from the tile origin: reads beyond
// return zero (used for the N=96 partial tile).
// This toolchain's builtin takes 6 args: (g0, g1, g2, g3, g4, cpol).
// ---------------------------------------------------------------------------
__device__ __forceinline__ void tdm_load_2d(unsigned int lds_off, const void* gptr,
                                            unsigned int tensor_d0, unsigned int tensor_d1,
                                            unsigned long long stride0_elems,
                                            unsigned int tile_d0, unsigned int tile_d1,
                                            unsigned int flags_w0)
{
    const unsigned long long ga = (unsigned long long)(uintptr_t)gptr;
    u32x4_t g0;
    g0[0] = 1u;                                        // count=1, user mode, no gather
    g0[1] = lds_off;                                   // LDS byte address
    g0[2] = (unsigned int)ga;                          // global_addr[31:0]
    g0[3] = (unsigned int)(ga >> 32) | (2u << 30);     // global_addr[56:32] | type=2
    i32x8_t g1;
    g1[0] = (int)flags_w0;                             // wg_mask=0 | data_size | pad cfg
    g1[1] = (int)((tensor_d0 & 0xFFFFu) << 16);        // [63:48]=tensor_dim0 lo
    g1[2] = (int)(((tensor_d0 >> 16) & 0xFFFFu) | ((tensor_d1 & 0xFFFFu) << 16));
    g1[3] = (int)(((tensor_d1 >> 16) & 0xFFFFu) | (tile_d0 << 16));
    g1[4] = (int)(tile_d1 & 0xFFFFu);                  // tile_dim1 | tile_dim2=0
    g1[5] = (int)(unsigned int)(stride0_elems & 0xFFFFFFFFull);
    g1[6] = (int)(unsigned int)((stride0_elems >> 32) & 0xFFFFull); // dim1_stride=0
    g1[7] = 0;
    i32x4_t z4 = {0, 0, 0, 0};
    i32x8_t z8 = {0, 0, 0, 0, 0, 0, 0, 0};
    __builtin_amdgcn_tensor_load_to_lds(g0, g1, z4, z4, z8, 0);
}
#endif

// ---------------------------------------------------------------------------
// LayerNorm: one 256-thread block per row of 1024
// ---------------------------------------------------------------------------
__global__ __launch_bounds__(256)
void layernorm_kernel(const float* __restrict__ x, const float* __restrict__ gamma,
                      const float* __restrict__ beta, float* __restrict__ out)
{
    __shared__ float rs[256], rs2[256];
    const int row = blockIdx.x;
    const float* xr = x + (size_t)row * 1024;
    float v[4];
    float s = 0.f, s2 = 0.f;
#pragma unroll
    for (int i = 0; i < 4; ++i) {
        v[i] = xr[threadIdx.x + i * 256];
        s += v[i];
        s2 += v[i] * v[i];
    }
    rs[threadIdx.x] = s;
    rs2[threadIdx.x] = s2;
    __syncthreads();
    for (int off = 128; off > 0; off >>= 1) {
        if (threadIdx.x < off) {
            rs[threadIdx.x]  += rs[threadIdx.x + off];
            rs2[threadIdx.x] += rs2[threadIdx.x + off];
        }
        __syncthreads();
    }
    const float mean = rs[0] * (1.f / 1024.f);
    const float var  = rs2[0] * (1.f / 1024.f) - mean * mean;
    const float rstd = rsqrtf(var + 1e-6f);
    float* orow = out + (size_t)row * 1024;
#pragma unroll
    for (int i = 0; i < 4; ++i) {
        const int c = threadIdx.x + i * 256;
        orow[c] = (v[i] - mean) * rstd * gamma[c] + beta[c];
    }
}

// ---------------------------------------------------------------------------
// fp32 WMMA GEMM: C[M,N] = A[M,K] @ B[K,N] + bias, optional softplus epilogue.
// Block: 256 threads = 8 waves (4 in M x 2 in N), wave tile 32x32 (2x2 frags),
// block tile 128x64. K staged through double-buffered LDS in chunks of 32 via
// the Tensor Data Mover (wave 0 issues DMA; DMA(i+1) overlaps compute(i)).
// As rows are padded to 33 floats (TDM pad: 1 DWORD after every 32) so column
// reads are bank-conflict free.  EPI: 0 = identity, 1 = softplus
// ---------------------------------------------------------------------------
template <int EPI>
__global__ __launch_bounds__(256)
void gemm_wmma(const float* __restrict__ A, const float* __restrict__ B,
               const float* __restrict__ bias, float* __restrict__ C,
               int M, int N, int K, int lda, int ldb, int ldc)
{
    __shared__ float As[2][BM][KC + 1];   // +1 pad (TDM pad feature / manual)
    __shared__ float Bs[2][KC][BN];

    const int tid  = threadIdx.x;
    const int lane = tid & 31;
    const int w    = tid >> 5;   // 0..7
    const int wm   = w & 3;      // 0..3  (M direction)
    const int wn   = w >> 2;     // 0..1  (N direction)
    const int half = lane >> 4;  // 0/1
    const int l16  = lane & 15;

    const int m0 = blockIdx.y * BM;
    const int n0 = blockIdx.x * BN;

    v8f_t c00 = {}, c01 = {}, c10 = {}, c11 = {};

    const int nchunks = K / KC;

#if USE_TDM
    // data_size=4B (=2); A adds pad_enable, pad_interval=32 DWORDs (=4),
    // pad_amount=1 DWORD (=0)
    const unsigned int FL_B = (2u << 16);
    const unsigned int FL_A = (2u << 16) | (1u << 20) | (4u << 22);

    auto issue_tiles = [&](int buf, int k0) {
        tdm_load_2d((unsigned int)(uintptr_t)&As[buf][0][0],
                    A + (size_t)m0 * lda + k0,
                    (unsigned int)(K - k0), (unsigned int)(M - m0),
                    (unsigned long long)lda, KC, BM, FL_A);
        tdm_load_2d((unsigned int)(uintptr_t)&Bs[buf][0][0],
                    B + (size_t)k0 * ldb + n0,
                    (unsigned int)(N - n0), (unsigned int)(K - k0),
                    (unsigned long long)ldb, BN, KC, FL_B);
    };

    if (w == 0) issue_tiles(0, 0);
#endif

    for (int ci = 0; ci < nchunks; ++ci) {
        const int buf = ci & 1;
#if USE_TDM
        if (w == 0) __builtin_amdgcn_s_wait_tensorcnt(0);
        __syncthreads();                      // tile ready AND prev readers done
        if (w == 0 && (ci + 1) < nchunks) issue_tiles(buf ^ 1, (ci + 1) * KC);
#else
        const int k0 = ci * KC;
#pragma unroll
        for (int it = 0; it < 4; ++it) {
            const int i  = it * 256 + tid;
            const int r  = i >> 3;
            const int c4 = (i & 7) << 2;
            const float4 vA = *(const float4*)(A + (size_t)(m0 + r) * lda + k0 + c4);
            As[buf][r][c4 + 0] = vA.x;
            As[buf][r][c4 + 1] = vA.y;
            As[buf][r][c4 + 2] = vA.z;
            As[buf][r][c4 + 3] = vA.w;
        }
#pragma unroll
        for (int it = 0; it < 8; ++it) {
            const int i  = it * 256 + tid;
            const int r  = i >> 6;
            const int c  = i & 63;
            const int gn = n0 + c;
            Bs[buf][r][c] = (gn < N) ? B[(size_t)(k0 + r) * ldb + gn] : 0.f;
        }
        __syncthreads();
#endif

#pragma unroll
        for (int ks = 0; ks < KC / 4; ++ks) {
            const int kb = ks * 4 + half * 2;  // this half-wave's K pair
            const int ra = wm * 32 + l16;
            const int cb = wn * 32 + l16;
            v2f_t a0, a1, b0, b1;
            a0[0] = As[buf][ra][kb];        a0[1] = As[buf][ra][kb + 1];
            a1[0] = As[buf][ra + 16][kb];   a1[1] = As[buf][ra + 16][kb + 1];
            b0[0] = Bs[buf][kb][cb];        b0[1] = Bs[buf][kb + 1][cb];
            b1[0] = Bs[buf][kb][cb + 16];   b1[1] = Bs[buf][kb + 1][cb + 16];
            c00 = __builtin_amdgcn_wmma_f32_16x16x4_f32(false, a0, false, b0, (short)0, c00, false, false);
            c01 = __builtin_amdgcn_wmma_f32_16x16x4_f32(false, a0, false, b1, (short)0, c01, false, false);
            c10 = __builtin_amdgcn_wmma_f32_16x16x4_f32(false, a1, false, b0, (short)0, c10, false, false);
            c11 = __builtin_amdgcn_wmma_f32_16x16x4_f32(false, a1, false, b1, (short)0, c11, false, false);
        }
#if !USE_TDM
        __syncthreads();
#endif
    }

    // --- store: bias + epilogue. C/D layout: VGPR r -> row r (lanes 0-15)
    //     or row r+8 (lanes 16-31); lane%16 -> column.
    auto store_frag = [&](const v8f_t& cf, int mb, int nb) {
        const int col = nb + l16;
        if (col >= N) return;
        const float bz = bias[col];
#pragma unroll
        for (int r = 0; r < 8; ++r) {
            const int row = mb + half * 8 + r;
            float vv = cf[r] + bz;
            if (EPI == 1) vv = softplusf(vv);
            C[(size_t)row * ldc + col] = vv;
        }
    };
    store_frag(c00, m0 + wm * 32,      n0 + wn * 32);
    store_frag(c01, m0 + wm * 32,      n0 + wn * 32 + 16);
    store_frag(c10, m0 + wm * 32 + 16, n0 + wn * 32);
    store_frag(c11, m0 + wm * 32 + 16, n0 + wn * 32 + 16);
}

// ---------------------------------------------------------------------------
// Selective scan: one thread per (b, d); 16-state recurrence over L=512.
// B/C vectors (dbc cols 64..95) are wave-uniform -> scalar-cached loads.
// y = sum_n h[n]*C[n] + u*D
// ---------------------------------------------------------------------------
__global__ __launch_bounds__(256)
void ssm_scan_kernel(const float* __restrict__ delta, const float* __restrict__ u,
                     const float* __restrict__ dbc, const float* __restrict__ A_log,
                     const float* __restrict__ Dp, float* __restrict__ y)
{
    const int d = blockIdx.x * 256 + threadIdx.x;  // 0..1023
    const int b = blockIdx.y;                       // 0..3

    float Arow[16];
#pragma unroll
    for (int n = 0; n < 16; ++n) Arow[n] = -__expf(A_log[d * 16 + n]);
    float h[16];
#pragma unroll
    for (int n = 0; n < 16; ++n) h[n] = 0.f;
    const float Dd = Dp[d];

    const size_t base = (size_t)b * 512;
    for (int t = 0; t < 512; ++t) {
        const size_t rd = (base + t) * 1024 + d;
        const float dl = delta[rd];
        const float uu = u[rd];
        const float* bc = dbc + (base + t) * 96;   // [64:80]=B, [80:96]=C
        const float dlu = dl * uu;
        float yv = 0.f;
#pragma unroll
        for (int n = 0; n < 16; ++n) {
            const float dA = __expf(dl * Arow[n]);
            h[n] = dA * h[n] + dlu * bc[64 + n];
            yv += h[n] * bc[80 + n];
        }
        y[rd] = yv + uu * Dd;
    }
}

// ---------------------------------------------------------------------------
// out = (y1 + y2) * silu(proj) + skip
// ---------------------------------------------------------------------------
__global__ __launch_bounds__(256)
void combine_kernel(const float* __restrict__ y1, const float* __restrict__ y2,
                    const float* __restrict__ proj, const float* __restrict__ skip,
                    float* __restrict__ out, int total)
{
    const int i = blockIdx.x * 256 + threadIdx.x;
    if (i >= total) return;
    const float p = proj[i];
    const float z = p / (1.f + __expf(-p));       // silu
    out[i] = (y1[i] + y2[i]) * z + skip[i];
}

// ---------------------------------------------------------------------------
extern "C" void kernel_launch(void* const* d_in, const int* in_sizes, int n_in,
                              void* d_out, int out_size, void* d_ws, size_t ws_size,
                              hipStream_t stream)
{
    (void)in_sizes; (void)n_in; (void)out_size; (void)ws_size;

    const float* inputs = (const float*)d_in[0];
    const float* gamma  = (const float*)d_in[1];
    const float* beta   = (const float*)d_in[2];
    const float* W_proj = (const float*)d_in[3];
    const float* b_proj = (const float*)d_in[4];
    const float* W_fc   = (const float*)d_in[5];
    const float* b_fc   = (const float*)d_in[6];
    const float* W_bc   = (const float*)d_in[7];
    const float* b_bc   = (const float*)d_in[8];
    const float* W_x1   = (const float*)d_in[9];
    const float* b_x1   = (const float*)d_in[10];
    const float* W_dt1  = (const float*)d_in[11];
    const float* b_dt1  = (const float*)d_in[12];
    const float* A_log1 = (const float*)d_in[13];
    const float* D1     = (const float*)d_in[14];
    const float* W_x2   = (const float*)d_in[15];
    const float* b_x2   = (const float*)d_in[16];
    const float* W_dt2  = (const float*)d_in[17];
    const float* b_dt2  = (const float*)d_in[18];
    const float* A_log2 = (const float*)d_in[19];
    const float* D2     = (const float*)d_in[20];

    const int ML = 2048;             // B*L
    const int D  = 1024;
    const int RN = 96;               // r + 2n
    const size_t SZ = (size_t)ML * D;

    float* ws   = (float*)d_ws;
    float* x_ln = ws;                // dead after proj GEMM; reused as y1
    float* proj = ws + 1 * SZ;
    float* u1   = ws + 2 * SZ;
    float* u2   = ws + 3 * SZ;
    float* dlt1 = ws + 4 * SZ;
    float* dlt2 = ws + 5 * SZ;
    float* y2   = ws + 6 * SZ;
    float* dbc1 = ws + 7 * SZ;
    float* dbc2 = dbc1 + (size_t)ML * RN;
    float* y1   = x_ln;

    // 1) LayerNorm
    layernorm_kernel<<<ML, 256, 0, stream>>>(inputs, gamma, beta, x_ln);

    // 2) proj = x_ln @ W_proj + b_proj
    dim3 gP(D / BN, ML / BM);        // (16, 16)
    gemm_wmma<0><<<gP, 256, 0, stream>>>(x_ln, W_proj, b_proj, proj, ML, D, D, D, D, D);

    // 3) u1 = softplus(proj @ W_fc + b_fc);  u2 = softplus(proj @ W_bc + b_bc)
    gemm_wmma<1><<<gP, 256, 0, stream>>>(proj, W_fc, b_fc, u1, ML, D, D, D, D, D);
    gemm_wmma<1><<<gP, 256, 0, stream>>>(proj, W_bc, b_bc, u2, ML, D, D, D, D, D);

    // 4) dbc = u @ W_x + b_x   (N = 96)
    dim3 gX((RN + BN - 1) / BN, ML / BM);   // (2, 16)
    gemm_wmma<0><<<gX, 256, 0, stream>>>(u1, W_x1, b_x1, dbc1, ML, RN, D, D, RN, RN);
    gemm_wmma<0><<<gX, 256, 0, stream>>>(u2, W_x2, b_x2, dbc2, ML, RN, D, D, RN, RN);

    // 5) delta = softplus(dbc[:, :64] @ W_dt + b_dt)   (K = 64, lda = 96)
    gemm_wmma<1><<<gP, 256, 0, stream>>>(dbc1, W_dt1, b_dt1, dlt1, ML, D, 64, RN, D, D);
    gemm_wmma<1><<<gP, 256, 0, stream>>>(dbc2, W_dt2, b_dt2, dlt2, ML, D, 64, RN, D, D);

    // 6) selective scans
    dim3 gS(D / 256, 4);             // (4, 4)
    ssm_scan_kernel<<<gS, 256, 0, stream>>>(dlt1, u1, dbc1, A_log1, D1, y1);
    ssm_scan_kernel<<<gS, 256, 0, stream>>>(dlt2, u2, dbc2, A_log2, D2, y2);

    // 7) combine: out = (y1 + y2) * silu(proj) + skip
    const int total = (int)SZ;
    combine_kernel<<<(total + 255) / 256, 256, 0, stream>>>(y1, y2, proj, inputs,
                                                            (float*)d_out, total);
}